// VorRec_37391985279582
// MI455X (gfx1250) — compile-verified
//
#include <hip/hip_runtime.h>
#include <math.h>

#define NUM_USERS 100000
#define NUM_ITEMS 20000
#define DIM       64
#define NUM_TAG   64
#define NUM_NEG   8
#define BTRIP     262144
#define M_IMP     2000
#define MARGIN1   0.1f
#define MARGIN2   0.1f
#define TAU       0.2f
#define EPS_C     1e-7f
#define MAX_NORM  (1.0f - 1e-5f)

typedef __attribute__((ext_vector_type(2))) float v2f;
typedef __attribute__((ext_vector_type(8))) float v8f;

__device__ __forceinline__ float artanh_c(float x) {
    x = fminf(fmaxf(x, -MAX_NORM), MAX_NORM);
    return 0.5f * (log1pf(x) - log1pf(-x));
}
__device__ __forceinline__ float acosh_c(float z) {
    z = fmaxf(z, 1.0f + EPS_C);
    return logf(z + sqrtf(z * z - 1.0f));
}

// ---------------------------------------------------------------------------
// K1: rank loss. One lane per triple; 10 row-gathers (float4) from the
// L2-resident embedding table, 9 dot products, deterministic block reduction.
// ---------------------------------------------------------------------------
__global__ void __launch_bounds__(256) k_rank(const float* __restrict__ emb,
                                              const int* __restrict__ trip,
                                              float* __restrict__ partials) {
    __shared__ float red[256];
    const int tid = threadIdx.x;
    const int g   = blockIdx.x * 256 + tid;
    const int* tp = trip + (size_t)g * (2 + NUM_NEG);

    const float4* eu = (const float4*)(emb + (size_t)tp[0] * DIM);
    const float4* ep = (const float4*)(emb + (size_t)tp[1] * DIM);
    const float4* en[NUM_NEG];
#pragma unroll
    for (int j = 0; j < NUM_NEG; ++j)
        en[j] = (const float4*)(emb + (size_t)tp[2 + j] * DIM);

    float dotp = 0.0f;
    float dotn[NUM_NEG];
#pragma unroll
    for (int j = 0; j < NUM_NEG; ++j) dotn[j] = 0.0f;
    float u0 = 0.0f, p0 = 0.0f, n0[NUM_NEG];

#pragma unroll
    for (int c = 0; c < DIM / 4; ++c) {
        float4 a = eu[c];
        float4 b = ep[c];
        if (c == 0) { u0 = a.x; p0 = b.x; }
        dotp += a.x * b.x + a.y * b.y + a.z * b.z + a.w * b.w;
#pragma unroll
        for (int j = 0; j < NUM_NEG; ++j) {
            float4 nb = en[j][c];
            if (c == 0) n0[j] = nb.x;
            dotn[j] += a.x * nb.x + a.y * nb.y + a.z * nb.z + a.w * nb.w;
        }
    }

    // lorentz inner = sum_{1:}(x*y) - x0*y0 = fulldot - 2*x0*y0
    float inner_p = dotp - 2.0f * u0 * p0;
    float dp      = acosh_c(-inner_p);
    float pos_d2  = dp * dp;
    float alpha   = acosh_c(u0);

    float contrib = 0.0f;
#pragma unroll
    for (int j = 0; j < NUM_NEG; ++j) {
        float inner_n = dotn[j] - 2.0f * u0 * n0[j];
        float dn      = acosh_c(-inner_n);
        contrib += fmaxf((pos_d2 - dn * dn + MARGIN2) * alpha, 0.0f);
    }

    red[tid] = contrib;
    __syncthreads();
    for (int off = 128; off > 0; off >>= 1) {
        if (tid < off) red[tid] += red[tid + off];
        __syncthreads();
    }
    if (tid == 0) partials[blockIdx.x] = red[0];
}

// ---------------------------------------------------------------------------
// K2a: precompute bt = logmap0(emb_tag), ||bt||^2, exp(-tag_weight).
// ---------------------------------------------------------------------------
__global__ void k_prep_tags(const float* __restrict__ emb_tag,
                            const float* __restrict__ tag_weight,
                            float* __restrict__ bt,
                            float* __restrict__ btn2,
                            float* __restrict__ inv_ew) {
    const int t = threadIdx.x; // 64 threads
    const float* row = emb_tag + (size_t)t * DIM;
    float s = 0.0f;
    for (int k = 0; k < DIM; ++k) s += row[k] * row[k];
    float n  = sqrtf(fmaxf(s, 1e-15f));
    float sc = artanh_c(n) / n;
    float s2 = 0.0f;
    for (int k = 0; k < DIM; ++k) {
        float v = sc * row[k];
        bt[(size_t)t * DIM + k] = v;
        s2 += v * v;
    }
    btn2[t]   = s2;
    inv_ew[t] = expf(-tag_weight[t]);
}

// ---------------------------------------------------------------------------
// K2b: classification loss. One workgroup (4 waves) per 16 items.
// Two f32 GEMMs via V_WMMA_F32_16X16X4_F32 (each wave owns an N-tile of 16),
// Mobius row ops in between, then masked hinge double-sum.
// ---------------------------------------------------------------------------
__global__ void __launch_bounds__(128) k_cls(const float* __restrict__ vtg,
                                             const float* __restrict__ linw,
                                             const float* __restrict__ linb,
                                             const int*   __restrict__ labels,
                                             const float* __restrict__ bt,
                                             const float* __restrict__ btn2,
                                             const float* __restrict__ inv_ew,
                                             float* __restrict__ partials) {
    __shared__ float sW [DIM][DIM + 1];   // lin_weight  [n][k]
    __shared__ float sBT[DIM][DIM + 1];   // bt          [t][k]
    __shared__ float sX [16][DIM + 1];    // enc, later 'a'
    __shared__ float sM [16][DIM + 1];    // mx, later dots
    __shared__ float sLogp[16][NUM_TAG];
    __shared__ float rowXn[16], rowAn2[16];
    __shared__ float sBias[DIM];
    __shared__ float sB2;
    __shared__ float red[128];

    const int tid  = threadIdx.x;
    const int lane = tid & 31;
    const int wid  = tid >> 5;
    const int i0   = blockIdx.x * 16;
    const int lh   = lane >> 4;   // lane half: selects K/M offset
    const int lm   = lane & 15;   // position within 16 (M for A, N for B/D)

    for (int idx = tid; idx < DIM * DIM; idx += 128) {
        int r = idx >> 6, c = idx & 63;
        sW[r][c]  = linw[idx];
        sBT[r][c] = bt[idx];
    }
    if (tid < DIM) sBias[tid] = linb[tid];
    for (int idx = tid; idx < 16 * DIM; idx += 128) {
        int r = idx >> 6, c = idx & 63;
        sX[r][c] = vtg[(size_t)(i0 + r) * DIM + c];
    }
    __syncthreads();

    if (tid == 0) {
        float s = 0.0f;
        for (int k = 0; k < DIM; ++k) s += sBias[k] * sBias[k];
        sB2 = s;
    }
    // enc = ball_projx(vtg rows); record xn = ||enc||
    if (tid < 16) {
        const int r = tid;
        float s = 0.0f;
        for (int k = 0; k < DIM; ++k) s += sX[r][k] * sX[r][k];
        float n  = sqrtf(fmaxf(s, 1e-15f));
        float sc = (n > MAX_NORM) ? (MAX_NORM / n) : 1.0f;
        float s2 = 0.0f;
        for (int k = 0; k < DIM; ++k) {
            float v = sX[r][k] * sc;
            sX[r][k] = v;
            s2 += v * v;
        }
        rowXn[r] = sqrtf(fmaxf(s2, 1e-15f));
    }
    __syncthreads();

    // GEMM1: mx[16x64] = enc @ W^T   (mx[m,n] = sum_k enc[m,k]*W[n,k])
    {
        const int n0 = wid * 16;
        v8f acc = {};
        for (int k0 = 0; k0 < DIM; k0 += 4) {
            v2f a, b;
            a.x = sX[lm][k0 + 2 * lh + 0];
            a.y = sX[lm][k0 + 2 * lh + 1];
            b.x = sW[n0 + lm][k0 + 2 * lh + 0];
            b.y = sW[n0 + lm][k0 + 2 * lh + 1];
            acc = __builtin_amdgcn_wmma_f32_16x16x4_f32(
                false, a, false, b, (short)0, acc, false, false);
        }
#pragma unroll
        for (int v = 0; v < 8; ++v) sM[v + 8 * lh][n0 + lm] = acc[v];
    }
    __syncthreads();

    // Per-row: finish mobius_matvec, mobius_add(bias), projx, logmap0 -> a
    if (tid < 16) {
        const int r = tid;
        float xn = rowXn[r];
        float s = 0.0f;
        for (int k = 0; k < DIM; ++k) s += sM[r][k] * sM[r][k];
        float mn  = sqrtf(fmaxf(s, 1e-15f));
        float fac = tanhf(mn / xn * artanh_c(xn)) / mn;
        float x2 = 0.0f, xy = 0.0f;
        for (int k = 0; k < DIM; ++k) {
            float h = fac * sM[r][k];
            sM[r][k] = h;
            x2 += h * h;
            xy += h * sBias[k];
        }
        float y2  = sB2;
        float den = fmaxf(1.0f + 2.0f * xy + x2 * y2, 1e-15f);
        float cx  = (1.0f + 2.0f * xy + y2) / den;
        float cy  = (1.0f - x2) / den;
        float s2  = 0.0f;
        for (int k = 0; k < DIM; ++k) {
            float h2 = cx * sM[r][k] + cy * sBias[k];
            sM[r][k] = h2;
            s2 += h2 * h2;
        }
        float n2  = sqrtf(fmaxf(s2, 1e-15f));
        float sc  = (n2 > MAX_NORM) ? (MAX_NORM / n2) : 1.0f;
        float n3  = sqrtf(fmaxf(sc * sc * s2, 1e-15f));
        float lsc = artanh_c(n3) / n3 * sc;
        float an2 = 0.0f;
        for (int k = 0; k < DIM; ++k) {
            float av = lsc * sM[r][k];
            sX[r][k] = av;           // 'a' rows overwrite enc
            an2 += av * av;
        }
        rowAn2[r] = an2;
    }
    __syncthreads();

    // GEMM2: dots[16x64] = a @ bt^T
    {
        const int n0 = wid * 16;
        v8f acc = {};
        for (int k0 = 0; k0 < DIM; k0 += 4) {
            v2f a, b;
            a.x = sX[lm][k0 + 2 * lh + 0];
            a.y = sX[lm][k0 + 2 * lh + 1];
            b.x = sBT[n0 + lm][k0 + 2 * lh + 0];
            b.y = sBT[n0 + lm][k0 + 2 * lh + 1];
            acc = __builtin_amdgcn_wmma_f32_16x16x4_f32(
                false, a, false, b, (short)0, acc, false, false);
        }
#pragma unroll
        for (int v = 0; v < 8; ++v) sM[v + 8 * lh][n0 + lm] = acc[v];
    }
    __syncthreads();

    // logp[r][t] = log( sqrt(max(||a||^2 + ||bt||^2 - 2*dot, 1e-12)) * e^{-w_t} )
    for (int idx = tid; idx < 16 * NUM_TAG; idx += 128) {
        int r = idx >> 6, t = idx & 63;
        float sq   = rowAn2[r] + btn2[t] - 2.0f * sM[r][t];
        float dist = sqrtf(fmaxf(sq, 1e-12f));
        sLogp[r][t] = logf(dist * inv_ew[t]);
    }
    __syncthreads();

    // masked hinge double-sum
    float acc = 0.0f;
    for (int idx = tid; idx < 16 * NUM_TAG; idx += 128) {
        int r = idx >> 6, j = idx & 63;
        if (labels[(size_t)(i0 + r) * NUM_TAG + j] > 0) {
            float lj = sLogp[r][j];
            float s = 0.0f;
            for (int k = 0; k < NUM_TAG; ++k)
                s += fmaxf(lj - sLogp[r][k] + MARGIN1, 0.0f);
            acc += s;
        }
    }
    red[tid] = acc;
    __syncthreads();
    for (int off = 64; off > 0; off >>= 1) {
        if (tid < off) red[tid] += red[tid + off];
        __syncthreads();
    }
    if (tid == 0) partials[blockIdx.x] = red[0];
}

// ---------------------------------------------------------------------------
// K3: contrastive (implication) loss. One 64-thread block per row; thread t
// computes ball_dist2(q, emb_tag[t]); dist_pos == d[b]; logsumexp over 65.
// ---------------------------------------------------------------------------
__global__ void __launch_bounds__(64) k_cl(const float* __restrict__ emb_tag,
                                           const int* __restrict__ implication,
                                           float* __restrict__ partials) {
    __shared__ float sT[NUM_TAG][DIM + 1];
    __shared__ float sd[NUM_TAG];
    const int tid = threadIdx.x;
    const int row = blockIdx.x;

    for (int idx = tid; idx < NUM_TAG * DIM; idx += 64)
        sT[idx >> 6][idx & 63] = emb_tag[idx];
    __syncthreads();

    const int ai = implication[row * 2 + 0];
    const int bi = implication[row * 2 + 1];

    // ball_dist2(q, y) with q = sT[ai], y = sT[tid]; mobius_add(-q, y)
    float q2 = 0.0f, y2 = 0.0f, qy = 0.0f;
    for (int k = 0; k < DIM; ++k) {
        float qk = sT[ai][k], yk = sT[tid][k];
        q2 += qk * qk;
        y2 += yk * yk;
        qy += qk * yk;
    }
    float xy  = -qy;
    float den = fmaxf(1.0f + 2.0f * xy + q2 * y2, 1e-15f);
    float cq  = -(1.0f + 2.0f * xy + y2) / den;
    float cy  = (1.0f - q2) / den;
    float s = 0.0f;
    for (int k = 0; k < DIM; ++k) {
        float v = cq * sT[ai][k] + cy * sT[tid][k];
        s += v * v;
    }
    float n = sqrtf(fmaxf(s, 1e-15f));
    float d = 2.0f * artanh_c(n);
    sd[tid] = d * d;
    __syncthreads();

    if (tid == 0) {
        float db = sd[bi];
        float vp = (db > 1e-9f) ? (-db / TAU) : -1e30f;
        float m = vp;
        for (int t = 0; t < NUM_TAG; ++t) {
            float v = (sd[t] > 1e-9f) ? (-sd[t] / TAU) : -1e30f;
            m = fmaxf(m, v);
        }
        float ssum = expf(vp - m);
        for (int t = 0; t < NUM_TAG; ++t) {
            float v = (sd[t] > 1e-9f) ? (-sd[t] / TAU) : -1e30f;
            ssum += expf(v - m);
        }
        partials[row] = m + logf(ssum) + db / TAU;
    }
}

// ---------------------------------------------------------------------------
// K4: deterministic final reduction (fixed strided order, single block).
// ---------------------------------------------------------------------------
__global__ void __launch_bounds__(256) k_final(const float* __restrict__ rank_p, int n1,
                                               const float* __restrict__ cls_p, int n2,
                                               const float* __restrict__ cl_p,  int n3,
                                               float* __restrict__ out) {
    __shared__ float red[256];
    const int tid = threadIdx.x;
    float a = 0.0f;
    for (int i = tid; i < n1; i += 256) a += rank_p[i];
    for (int i = tid; i < n2; i += 256) a += cls_p[i];
    for (int i = tid; i < n3; i += 256) a += cl_p[i];
    red[tid] = a;
    __syncthreads();
    for (int off = 128; off > 0; off >>= 1) {
        if (tid < off) red[tid] += red[tid + off];
        __syncthreads();
    }
    if (tid == 0) out[0] = red[0];
}

extern "C" void kernel_launch(void* const* d_in, const int* in_sizes, int n_in,
                              void* d_out, int out_size, void* d_ws, size_t ws_size,
                              hipStream_t stream) {
    (void)in_sizes; (void)n_in; (void)out_size; (void)ws_size;
    const float* emb      = (const float*)d_in[0];
    const float* vtg      = (const float*)d_in[1];
    const float* emb_tag  = (const float*)d_in[2];
    const float* linw     = (const float*)d_in[3];
    const float* linb     = (const float*)d_in[4];
    const float* tagw     = (const float*)d_in[5];
    const int*   triples  = (const int*)d_in[6];
    const int*   labels   = (const int*)d_in[7];
    const int*   implic   = (const int*)d_in[8];

    float* ws     = (float*)d_ws;
    float* bt     = ws;                 // 4096
    float* btn2   = ws + 4096;          // 64
    float* inv_ew = ws + 4160;          // 64
    float* rank_p = ws + 4224;          // 1024
    float* cls_p  = rank_p + 1024;      // 1250
    float* cl_p   = cls_p + 1250;       // 2000

    const int NB_RANK = BTRIP / 256;        // 1024
    const int NB_CLS  = NUM_ITEMS / 16;     // 1250

    k_rank<<<NB_RANK, 256, 0, stream>>>(emb, triples, rank_p);
    k_prep_tags<<<1, 64, 0, stream>>>(emb_tag, tagw, bt, btn2, inv_ew);
    k_cls<<<NB_CLS, 128, 0, stream>>>(vtg, linw, linb, labels, bt, btn2, inv_ew, cls_p);
    k_cl<<<M_IMP, 64, 0, stream>>>(emb_tag, implic, cl_p);
    k_final<<<1, 256, 0, stream>>>(rank_p, NB_RANK, cls_p, NB_CLS, cl_p, M_IMP,
                                   (float*)d_out);
}